// KNNConv_35253091566107
// MI455X (gfx1250) — compile-verified
//
#include <hip/hip_runtime.h>
#include <hip/hip_bf16.h>

typedef __attribute__((ext_vector_type(2))) float v2f;
typedef __attribute__((ext_vector_type(8))) float v8f;

#define M_PTS 8192
#define CDIM  64
#define KNN   20
#define ODIM  64

// ---------------------------------------------------------------------------
// WMMA helpers (V_WMMA_F32_16X16X4_F32, wave32)
// A: 16xK tile, row = lane&15, VGPR0/1 hold k0/k1 (lanes 0-15) and k2/k3 (16-31)
// B: "16 output cols x K" tile in the same per-lane layout (B-matrix rows = K)
// C/D: lane<16 -> rows 0..7, lane>=16 -> rows 8..15; col = lane&15
// ---------------------------------------------------------------------------

// a, b: per-lane row pointers (already offset by lr*ld + k0). Loads a chunk of
// 8 A-pairs + 8 B-pairs into registers, then issues 8 WMMAs -> batched DS
// traffic (single dscnt wait per chunk) instead of load/wait/wmma lockstep.
__device__ __forceinline__ v8f wmma_chunks(const float* __restrict__ a,
                                           const float* __restrict__ b,
                                           int kdim, v8f acc) {
  #pragma unroll
  for (int k = 0; k < kdim; k += 32) {
    v2f av[8], bv[8];
    #pragma unroll
    for (int t = 0; t < 8; ++t) {
      av[t].x = a[k + t * 4];
      av[t].y = a[k + t * 4 + 1];
      bv[t].x = b[k + t * 4];
      bv[t].y = b[k + t * 4 + 1];
    }
    #pragma unroll
    for (int t = 0; t < 8; ++t)
      acc = __builtin_amdgcn_wmma_f32_16x16x4_f32(false, av[t], false, bv[t],
                                                  (short)0, acc, false, false);
  }
  return acc;
}

__device__ __forceinline__ void store_tile16(float* __restrict__ D, int ldd, v8f acc) {
  const int lane = threadIdx.x & 31;
  const int col  = lane & 15;
  const int rb   = (lane < 16) ? 0 : 8;
  #pragma unroll
  for (int v = 0; v < 8; ++v) D[(rb + v) * ldd + col] = acc[v];
}

// Async global -> LDS copy, 16B per lane, tracked by ASYNCcnt (inline asm:
// no clang builtin for GLOBAL_LOAD_ASYNC_TO_LDS_* on this toolchain).
__device__ __forceinline__ void async_copy_b128(unsigned lds_byte_addr,
                                                int gvoff_bytes,
                                                const void* gbase) {
  asm volatile("global_load_async_to_lds_b128 %0, %1, %2"
               :: "v"(lds_byte_addr), "v"(gvoff_bytes), "s"(gbase)
               : "memory");
}

// ---------------------------------------------------------------------------
// Kernel 0: squared norms of positions
// ---------------------------------------------------------------------------
__global__ __launch_bounds__(256) void norms_kernel(const float* __restrict__ pos,
                                                    float* __restrict__ sq) {
  int m = blockIdx.x * blockDim.x + threadIdx.x;
  if (m < M_PTS) {
    const float* p = pos + m * CDIM;
    float s = 0.f;
    #pragma unroll 8
    for (int c = 0; c < CDIM; ++c) { float v = p[c]; s += v * v; }
    sq[m] = s;
  }
}

// ---------------------------------------------------------------------------
// Kernel 1: fused distance-GEMM (WMMA f32) + per-row top-20 selection.
// 256 threads (8 waves), 64 query rows per block.
//  - A operand (query tile) register-resident for the whole kernel
//  - 4 independent accumulator chains per wave (DS batching + WMMA ILP)
//  - candidate tiles double-buffered via global_load_async_to_lds_b128
// ---------------------------------------------------------------------------
__global__ __launch_bounds__(256) void knn_topk_kernel(const float* __restrict__ pos,
                                                       const float* __restrict__ sq,
                                                       int* __restrict__ knn) {
  extern __shared__ float smem[];
  float* Qs  = smem;                   // 64  x 68
  float* Cs0 = Qs + 64 * 68;           // 128 x 68  (ping)
  float* Cs1 = Cs0 + 128 * 68;         // 128 x 68  (pong)
  float* Pb  = Cs1 + 128 * 68;         // 64  x 132
  float* sqq = Pb + 64 * 132;          // 64
  float* sqc = sqq + 64;               // 128

  const int tid    = threadIdx.x;
  const int wave   = tid >> 5;
  const int lane   = tid & 31;
  const int rowBlk = blockIdx.x * 64;

  // stage the 64 query rows (plain path; one-time)
  for (int i = tid; i < 64 * CDIM; i += 256) {
    int r = i >> 6, c = i & 63;
    Qs[r * 68 + c] = pos[(rowBlk + r) * CDIM + c];
  }
  if (tid < 64) sqq[tid] = sq[rowBlk + tid];

  // async prefetch of candidate tile 0 into Cs0 (overlaps with Q staging)
  {
    const unsigned ldsBase = (unsigned)(uintptr_t)(void*)Cs0;
    #pragma unroll
    for (int t = 0; t < 8; ++t) {
      const int chunk = tid + t * 256;        // 2048 x 16B = 32KB tile
      const int r = chunk >> 4, ci = chunk & 15;
      async_copy_b128(ldsBase + (unsigned)(r * 272 + ci * 16),
                      r * 256 + ci * 16, pos);
    }
  }

  __syncthreads();  // Qs visible (does not wait on ASYNCcnt)

  // hoist A (query rows of this wave's row tile) into registers: invariant
  const int rt     = wave >> 1;            // row tile owned by this wave
  const int ctBase = (wave & 1) * 4;       // col tiles ctBase..ctBase+3
  const int lr     = lane & 15;
  const int k0     = (lane < 16) ? 0 : 2;
  v2f areg[16];
  {
    const float* arow = Qs + (rt * 16 + lr) * 68 + k0;
    #pragma unroll
    for (int kb = 0; kb < 16; ++kb) {
      areg[kb].x = arow[kb * 4];
      areg[kb].y = arow[kb * 4 + 1];
    }
  }

  float bestD[KNN];
  int   bestI[KNN];
  #pragma unroll
  for (int j = 0; j < KNN; ++j) { bestD[j] = 3.0e38f; bestI[j] = -1; }

  const int myrow = tid >> 2;              // 0..63 (top-k phase)
  const int sub   = tid & 3;               // each scans a contiguous 32 cols

  for (int cb = 0; cb < M_PTS / 128; ++cb) {
    float* cur = (cb & 1) ? Cs1 : Cs0;
    float* nxt = (cb & 1) ? Cs0 : Cs1;

    // prefetch next tile, then wait for current tile's 8 loads to retire
    if (cb + 1 < M_PTS / 128) {
      const unsigned ldsBase = (unsigned)(uintptr_t)(void*)nxt;
      #pragma unroll
      for (int t = 0; t < 8; ++t) {
        const int chunk = tid + t * 256;
        const int r = chunk >> 4, ci = chunk & 15;
        async_copy_b128(ldsBase + (unsigned)(r * 272 + ci * 16),
                        ((cb + 1) * 128 + r) * 256 + ci * 16, pos);
      }
      asm volatile("s_wait_asynccnt 0x8" ::: "memory");
    } else {
      asm volatile("s_wait_asynccnt 0x0" ::: "memory");
    }
    if (tid < 128) sqc[tid] = sq[cb * 128 + tid];
    __syncthreads();   // publish DMA-written tile + sqc to all waves

    // 64x128 tile: 4 interleaved accumulator chains (4 col tiles) per wave
    {
      const float* brow0 = cur + ((ctBase + 0) * 16 + lr) * 68 + k0;
      const float* brow1 = brow0 + 16 * 68;
      const float* brow2 = brow0 + 32 * 68;
      const float* brow3 = brow0 + 48 * 68;
      v8f acc[4];
      acc[0] = v8f{}; acc[1] = v8f{}; acc[2] = v8f{}; acc[3] = v8f{};
      #pragma unroll
      for (int kb = 0; kb < 16; ++kb) {
        v2f b0, b1, b2, b3;
        b0.x = brow0[kb * 4]; b0.y = brow0[kb * 4 + 1];
        b1.x = brow1[kb * 4]; b1.y = brow1[kb * 4 + 1];
        b2.x = brow2[kb * 4]; b2.y = brow2[kb * 4 + 1];
        b3.x = brow3[kb * 4]; b3.y = brow3[kb * 4 + 1];
        acc[0] = __builtin_amdgcn_wmma_f32_16x16x4_f32(false, areg[kb], false, b0,
                                                       (short)0, acc[0], false, false);
        acc[1] = __builtin_amdgcn_wmma_f32_16x16x4_f32(false, areg[kb], false, b1,
                                                       (short)0, acc[1], false, false);
        acc[2] = __builtin_amdgcn_wmma_f32_16x16x4_f32(false, areg[kb], false, b2,
                                                       (short)0, acc[2], false, false);
        acc[3] = __builtin_amdgcn_wmma_f32_16x16x4_f32(false, areg[kb], false, b3,
                                                       (short)0, acc[3], false, false);
      }
      const int rb = rt * 16 + ((lane < 16) ? 0 : 8);
      #pragma unroll
      for (int j = 0; j < 4; ++j) {
        const int col = (ctBase + j) * 16 + lr;
        const float sc = sqc[col];
        #pragma unroll
        for (int v = 0; v < 8; ++v) {
          const int r = rb + v;
          Pb[r * 132 + col] = sqq[r] + sc - 2.0f * acc[j][v];
        }
      }
    }
    __syncthreads();

    // running top-20: 4 threads/row, contiguous 32-col segments, b128 reads
    {
      const float4* prow = (const float4*)(Pb + myrow * 132 + sub * 32);
      const int colBase = cb * 128 + sub * 32;
      #pragma unroll
      for (int j = 0; j < 8; ++j) {
        const float4 dv = prow[j];
        #pragma unroll
        for (int e = 0; e < 4; ++e) {
          const float d = (&dv.x)[e];
          if (d < bestD[KNN - 1]) {
            bestD[KNN - 1] = d;
            bestI[KNN - 1] = colBase + j * 4 + e;
            #pragma unroll
            for (int s = KNN - 1; s > 0; --s) {
              if (bestD[s] < bestD[s - 1]) {
                float td = bestD[s]; bestD[s] = bestD[s - 1]; bestD[s - 1] = td;
                int   ti = bestI[s]; bestI[s] = bestI[s - 1]; bestI[s - 1] = ti;
              }
            }
          }
        }
      }
    }
    __syncthreads();   // protect Pb + pong buffer before next iteration
  }

  // merge the 4 sorted partial lists per row (reuse Cs0/Cs1 region)
  float* mD = Cs0;                        // 64*80 floats
  int*   mI = (int*)(Cs0 + 64 * 80);      // 64*80 ints
  #pragma unroll
  for (int j = 0; j < KNN; ++j) {
    mD[(myrow * 4 + sub) * KNN + j] = bestD[j];
    mI[(myrow * 4 + sub) * KNN + j] = bestI[j];
  }
  __syncthreads();
  if (tid < 64) {
    const float* L = mD + tid * 80;
    const int*   I = mI + tid * 80;
    int p0 = 0, p1 = 0, p2 = 0, p3 = 0;
    for (int j = 0; j < KNN; ++j) {
      float d0 = (p0 < KNN) ? L[p0]      : 3.0e38f;
      float d1 = (p1 < KNN) ? L[20 + p1] : 3.0e38f;
      float d2 = (p2 < KNN) ? L[40 + p2] : 3.0e38f;
      float d3 = (p3 < KNN) ? L[60 + p3] : 3.0e38f;
      float dm = d0; int sel = 0;
      if (d1 < dm) { dm = d1; sel = 1; }
      if (d2 < dm) { dm = d2; sel = 2; }
      if (d3 < dm) { dm = d3; sel = 3; }
      int idx;
      if      (sel == 0) { idx = I[p0];      ++p0; }
      else if (sel == 1) { idx = I[20 + p1]; ++p1; }
      else if (sel == 2) { idx = I[40 + p2]; ++p2; }
      else               { idx = I[60 + p3]; ++p3; }
      knn[(rowBlk + tid) * KNN + j] = idx;
    }
  }
}

// ---------------------------------------------------------------------------
// Kernel 2: edge MLP (two WMMA f32 GEMMs + ReLU) + max over K
// block = 256 threads (8 waves), 8 points -> 160 edge rows = 10 row tiles
// ---------------------------------------------------------------------------
__global__ __launch_bounds__(256) void edge_mlp_kernel(const float* __restrict__ feat,
                                                       const int* __restrict__ knn,
                                                       const float* __restrict__ W1,
                                                       const float* __restrict__ b1,
                                                       const float* __restrict__ W2,
                                                       const float* __restrict__ b2,
                                                       float* __restrict__ out) {
  extern __shared__ float smem[];
  float* W1t  = smem;                 // 64 x 132  (W1 transposed: [o][k], k=0..127)
  float* W2t  = W1t + 64 * 132;       // 64 x 68   (W2 transposed: [o][k], k=0..63)
  float* b1s  = W2t + 64 * 68;        // 64
  float* b2s  = b1s + 64;             // 64
  float* Fcs  = b2s + 64;             // 8 x 68    center features
  float* Diff = Fcs + 8 * 68;         // 160 x 68  neighbor - center
  float* Hs   = Diff + 160 * 68;      // 160 x 68  relu(layer1)
  float* Gs   = Diff;                 // layer2 output aliases Diff

  const int tid  = threadIdx.x;
  const int wave = tid >> 5;
  const int lane = tid & 31;

  for (int i = tid; i < 128 * 64; i += 256) {
    int k = i >> 6, o = i & 63;
    W1t[o * 132 + k] = W1[i];
  }
  for (int i = tid; i < 64 * 64; i += 256) {
    int k = i >> 6, o = i & 63;
    W2t[o * 68 + k] = W2[i];
  }
  if (tid < 64) { b1s[tid] = b1[tid]; b2s[tid] = b2[tid]; }

  // center features (stored once per point)
  for (int i = tid; i < 8 * 64; i += 256) {
    int pl = i >> 6, c = i & 63;
    Fcs[pl * 68 + c] = feat[(blockIdx.x * 8 + pl) * CDIM + c];
  }
  // neighbor-minus-center rows
  if (tid < 160) {
    const int pl = tid / 20, k = tid % 20;
    const int point = blockIdx.x * 8 + pl;
    const int nbr   = knn[point * KNN + k];
    const float* fc = feat + point * CDIM;
    const float* fn = feat + nbr * CDIM;
    float* dr = Diff + tid * 68;
    #pragma unroll 4
    for (int c = 0; c < CDIM; ++c) dr[c] = fn[c] - fc[c];
  }
  __syncthreads();

  const int lr = lane & 15;
  const int k0 = (lane < 16) ? 0 : 2;

  // ---- layer 1: [160 x 128] @ [128 x 64] + b1, ReLU -> Hs
  #pragma unroll
  for (int t5 = 0; t5 < 5; ++t5) {
    const int tile = wave + t5 * 8;     // 40 tiles, 5 per wave
    const int rt = tile >> 2, ct = tile & 3;
    const int row = rt * 16 + lr;
    const int pl  = row / 20;
    const float* crow = Fcs + pl * 68 + k0;
    const float* drow = Diff + row * 68 + k0;
    const float* brow = W1t + (ct * 16 + lr) * 132 + k0;
    const float bv0 = b1s[ct * 16 + lr];
    v8f acc = {bv0, bv0, bv0, bv0, bv0, bv0, bv0, bv0};
    acc = wmma_chunks(crow, brow, 64, acc);        // center half (k 0..63)
    acc = wmma_chunks(drow, brow + 64, 64, acc);   // diff half (k 64..127)
    #pragma unroll
    for (int v = 0; v < 8; ++v) acc[v] = fmaxf(acc[v], 0.0f);
    store_tile16(Hs + rt * 16 * 68 + ct * 16, 68, acc);
  }
  __syncthreads();

  // ---- layer 2: [160 x 64] @ [64 x 64] + b2 -> Gs
  #pragma unroll
  for (int t5 = 0; t5 < 5; ++t5) {
    const int tile = wave + t5 * 8;
    const int rt = tile >> 2, ct = tile & 3;
    const float* arow = Hs + (rt * 16 + lr) * 68 + k0;
    const float* brow = W2t + (ct * 16 + lr) * 68 + k0;
    const float bv0 = b2s[ct * 16 + lr];
    v8f acc = {bv0, bv0, bv0, bv0, bv0, bv0, bv0, bv0};
    acc = wmma_chunks(arow, brow, 64, acc);
    store_tile16(Gs + rt * 16 * 68 + ct * 16, 68, acc);
  }
  __syncthreads();

  // ---- max over the 20 neighbors, write output
  for (int i = tid; i < 8 * ODIM; i += 256) {
    const int pl = i >> 6, o = i & 63;
    const float* g = Gs + (pl * 20) * 68 + o;
    float m = g[0];
    #pragma unroll 5
    for (int k = 1; k < KNN; ++k) m = fmaxf(m, g[k * 68]);
    out[(blockIdx.x * 8 + pl) * ODIM + o] = m;
  }
}

// ---------------------------------------------------------------------------
extern "C" void kernel_launch(void* const* d_in, const int* in_sizes, int n_in,
                              void* d_out, int out_size, void* d_ws, size_t ws_size,
                              hipStream_t stream) {
  const float* pos  = (const float*)d_in[0];
  const float* feat = (const float*)d_in[1];
  const float* W1   = (const float*)d_in[2];
  const float* b1   = (const float*)d_in[3];
  const float* W2   = (const float*)d_in[4];
  const float* b2   = (const float*)d_in[5];
  float* out = (float*)d_out;

  float* sq  = (float*)d_ws;                                        // M floats
  int*   knn = (int*)((char*)d_ws + (size_t)M_PTS * sizeof(float)); // M*K ints

  norms_kernel<<<M_PTS / 256, 256, 0, stream>>>(pos, sq);

  constexpr int SMEM_KNN =
      (64 * 68 + 2 * 128 * 68 + 64 * 132 + 64 + 128) * 4;  // ~121.5 KB
  knn_topk_kernel<<<M_PTS / 64, 256, SMEM_KNN, stream>>>(pos, sq, knn);

  constexpr int SMEM_MLP =
      (64 * 132 + 64 * 68 + 64 + 64 + 8 * 68 + 160 * 68 + 160 * 68) * 4;
  edge_mlp_kernel<<<M_PTS / 8, 256, SMEM_MLP, stream>>>(feat, knn, W1, b1, W2, b2, out);
}